// SCL_12927851560994
// MI455X (gfx1250) — compile-verified
//
#include <hip/hip_runtime.h>

// Cosine-similarity diagonal loss for MI455X (gfx1250, wave32).
//
// reference: loss = -(1/n) * sum_i dot(f1_i,f2_i) / (max(||f1_i||,eps)*max(||f2_i||,eps))
// n = 8192, d = 256, fp32.  Memory-bound: ~16.8 MB @ 23.3 TB/s ~= 0.72 us.
//
// Strategy: one wave per 16 rows. V_WMMA_F32_16X16X4_F32 accumulates
//   D12 = F1_tile x F2_tile^T   (diag = dot(f1_i, f2_i))
//   D11 = F1_tile x F1_tile^T   (diag = ||f1_i||^2)
//   D22 = F2_tile x F2_tile^T   (diag = ||f2_i||^2)
// For 32-bit data the A-operand register layout of X equals the B-operand
// layout of X^T, so a single v2f load per matrix per k-step feeds both slots.

typedef __attribute__((ext_vector_type(2))) float v2f;
typedef __attribute__((ext_vector_type(8))) float v8f;

#define DIM 256
#define TILE_ROWS 16
#define EPS 1e-8f

__global__ __launch_bounds__(32) void SCL_cosdiag_wmma(
    const float* __restrict__ f1, const float* __restrict__ f2,
    float* __restrict__ partial) {
  const int lane = threadIdx.x;        // 0..31, wave32
  const int r    = lane & 15;          // row within 16-row tile
  const int half = lane >> 4;          // 0: K = k0,k0+1 ; 1: K = k0+2,k0+3
  const int row0 = blockIdx.x * TILE_ROWS;

  const float* p1 = f1 + (size_t)(row0 + r) * DIM + 2 * half;
  const float* p2 = f2 + (size_t)(row0 + r) * DIM + 2 * half;

  v8f acc12 = {};  // f1 . f2^T
  v8f acc11 = {};  // f1 . f1^T
  v8f acc22 = {};  // f2 . f2^T

#pragma unroll 8
  for (int k = 0; k < DIM; k += 4) {
    // A-layout 16x4 f32: VGPR0 = K=k0 (lanes 0-15) / k0+2 (lanes 16-31),
    //                    VGPR1 = K=k0+1 / k0+3.  Identical to B-layout of the
    // transpose (N on lanes, K split the same way), so reuse for both operands.
    v2f a = *(const v2f*)(p1 + k);
    v2f b = *(const v2f*)(p2 + k);
    acc12 = __builtin_amdgcn_wmma_f32_16x16x4_f32(
        false, a, false, b, (short)0, acc12, false, false);
    acc11 = __builtin_amdgcn_wmma_f32_16x16x4_f32(
        false, a, false, a, (short)0, acc11, false, false);
    acc22 = __builtin_amdgcn_wmma_f32_16x16x4_f32(
        false, b, false, b, (short)0, acc22, false, false);
  }

  // Diagonal of 16x16 f32 C/D: (i,i) at lane i, VGPR i (i<8) or
  // lane i+16, VGPR i-8 (i>=8). Lanes 8..23 hold no diagonal element.
  int idx = (lane < 8) ? lane : (lane >= 24 ? lane - 24 : -1);

  float dotv = 0.0f, n1 = 0.0f, n2 = 0.0f;
#pragma unroll
  for (int v = 0; v < 8; ++v) {   // constant-index select tree, no movrel
    const bool m = (v == idx);
    dotv = m ? acc12[v] : dotv;
    n1   = m ? acc11[v] : n1;
    n2   = m ? acc22[v] : n2;
  }

  // Non-diagonal lanes: dotv=0 -> sim=0; harmless in the sum.
  const float norm1 = fmaxf(sqrtf(n1), EPS);
  const float norm2 = fmaxf(sqrtf(n2), EPS);
  float sim = dotv / (norm1 * norm2);

  // wave32 butterfly reduction over the 16 row-sims (zeros elsewhere)
#pragma unroll
  for (int off = 16; off >= 1; off >>= 1)
    sim += __shfl_xor(sim, off, 32);

  if (lane == 0) partial[blockIdx.x] = sim;
}

// Deterministic final reduction (fixed order, no float atomics).
__global__ __launch_bounds__(32) void SCL_cosdiag_finish(
    const float* __restrict__ partial, float* __restrict__ out,
    int nblocks, float inv_n) {
  const int lane = threadIdx.x;
  float s = 0.0f;
  for (int i = lane; i < nblocks; i += 32) s += partial[i];
#pragma unroll
  for (int off = 16; off >= 1; off >>= 1)
    s += __shfl_xor(s, off, 32);
  if (lane == 0) out[0] = -s * inv_n;
}

extern "C" void kernel_launch(void* const* d_in, const int* in_sizes, int n_in,
                              void* d_out, int out_size, void* d_ws, size_t ws_size,
                              hipStream_t stream) {
  const float* f1 = (const float*)d_in[0];
  const float* f2 = (const float*)d_in[1];
  float* out = (float*)d_out;
  float* partial = (float*)d_ws;              // nblocks floats of scratch

  const int total = in_sizes[0];              // n * d
  const int n = total / DIM;                  // 8192
  const int nblocks = n / TILE_ROWS;          // 512 one-wave workgroups

  SCL_cosdiag_wmma<<<nblocks, 32, 0, stream>>>(f1, f2, partial);
  SCL_cosdiag_finish<<<1, 32, 0, stream>>>(partial, out, nblocks,
                                           1.0f / (float)n);
}